// Dilation2D_82231443849911
// MI455X (gfx1250) — compile-verified
//
#include <hip/hip_runtime.h>
#include <cstdint>

// Problem constants (from setup_inputs): x:(8,512,512,32) f32, w:(4,4,32) f32.
#define B_   8
#define H_   512
#define W_   512
#define C_   32
#define KH_  4
#define KW_  4

// Tiling: one block = 32x8 output pixels, all 32 channels.
#define TILE_W     32
#define TILE_H     8
#define HALO_W     (TILE_W + 3)              // 35 (pad-left 1, pad-right 2)
#define HALO_H     (TILE_H + 3)              // 11
#define HALO_PIX   (HALO_W * HALO_H)         // 385 pixels
#define QPP        (C_ / 4)                  // 8 float4 quads per pixel
#define HALO_QUADS (HALO_PIX * QPP)          // 3080 float4 = 48.1 KB LDS
#define NTHREADS   256                       // 8 waves (wave32)

__device__ __forceinline__ float4 vmax4(float4 a, float4 b) {
  return make_float4(fmaxf(a.x, b.x), fmaxf(a.y, b.y),
                     fmaxf(a.z, b.z), fmaxf(a.w, b.w));
}
__device__ __forceinline__ float4 vadd4(float4 a, float4 b) {
  return make_float4(a.x + b.x, a.y + b.y, a.z + b.z, a.w + b.w);
}

// ---- CDNA5 async global->LDS copy (gfx1250-specific data path) ------------
// Probe round 1 revealed the builtin's signature: 4 args, first param is
// "vector_size(16) int __device__ *"  ==  v4i in AS(1) (HIP __device__ = global).
// So: (v4i as1* gsrc, v4i as3* ldst, imm offset, imm cpol).
__device__ __forceinline__ void async_copy_b128(const float* gp, float4* lp) {
#if __has_builtin(__builtin_amdgcn_global_load_async_to_lds_b128)
  typedef int v4i_t __attribute__((vector_size(16)));
  typedef __attribute__((address_space(1))) v4i_t* gvec_t;
  typedef __attribute__((address_space(3))) v4i_t* lvec_t;
  __builtin_amdgcn_global_load_async_to_lds_b128(
      (gvec_t)(float*)gp, (lvec_t)lp, 0, 0);
#else
  // Flat shared addresses carry the LDS offset in the low 32 bits (aperture
  // lives in the high dword), so truncation yields a valid LDS byte address.
  uint32_t la = (uint32_t)(uintptr_t)lp;
  uint64_t ga = (uint64_t)(uintptr_t)gp;
  asm volatile("global_load_async_to_lds_b128 %0, %1, off"
               :: "v"(la), "v"(ga) : "memory");
#endif
}

__device__ __forceinline__ void wait_async_all() {
#if __has_builtin(__builtin_amdgcn_s_wait_asynccnt)
  __builtin_amdgcn_s_wait_asynccnt(0);
#else
  asm volatile("s_wait_asynccnt 0" ::: "memory");
#endif
}

// ---- tap fetch from LDS halo; invalid taps become -inf (max identity) -----
__device__ __forceinline__ float4 load_tap(const float4* tile, int r, int xt,
                                           int c4, bool valid) {
  float4 v = tile[(r * HALO_W + xt) * QPP + c4];
  if (!valid) {
    const float ninf = -__builtin_inff();
    v = make_float4(ninf, ninf, ninf, ninf);
  }
  return v;
}

template <bool CHECKED>
__device__ __forceinline__ void dilate_body(const float4* tile,
                                            const float4 (&wq)[KH_][KW_],
                                            float* __restrict__ out,
                                            int b, int ty0, int tx0,
                                            int xo, int c4) {
  bool cval[KW_];
#pragma unroll
  for (int j = 0; j < KW_; ++j)
    cval[j] = !CHECKED || ((unsigned)(tx0 + xo - 1 + j) < (unsigned)W_);

  // Rolling 4-row x 4-tap register window: 4 ds_load_b128 per new output row.
  float4 win[4][KW_];
#pragma unroll
  for (int r = 0; r < 3; ++r) {
    const bool rv = !CHECKED || ((unsigned)(ty0 + r - 1) < (unsigned)H_);
#pragma unroll
    for (int j = 0; j < KW_; ++j)
      win[r][j] = load_tap(tile, r, xo + j, c4, rv && cval[j]);
  }

#pragma unroll
  for (int yo = 0; yo < TILE_H; ++yo) {
    const int rn = yo + 3;
    const bool rv = !CHECKED || ((unsigned)(ty0 + rn - 1) < (unsigned)H_);
#pragma unroll
    for (int j = 0; j < KW_; ++j)
      win[rn & 3][j] = load_tap(tile, rn, xo + j, c4, rv && cval[j]);

    const float ninf = -__builtin_inff();
    float4 acc = make_float4(ninf, ninf, ninf, ninf);
#pragma unroll
    for (int i = 0; i < KH_; ++i)
#pragma unroll
      for (int j = 0; j < KW_; ++j)
        acc = vmax4(acc, vadd4(win[(yo + i) & 3][j], wq[i][j]));

    float4* op = (float4*)(out +
        ((((size_t)b * H_) + (ty0 + yo)) * W_ + (tx0 + xo)) * C_ + 4 * c4);
    *op = acc;  // global_store_b128, fully coalesced per wave
  }
}

__global__ __launch_bounds__(NTHREADS)
void dilation2d_kernel(const float* __restrict__ x,
                       const float* __restrict__ w,
                       float* __restrict__ out) {
  __shared__ float4 tile[HALO_QUADS];

  const int tx0 = blockIdx.x * TILE_W;
  const int ty0 = blockIdx.y * TILE_H;
  const int b   = blockIdx.z;
  const int tid = threadIdx.x;

  // ---- stage halo tile into LDS with async b128 copies (ASYNCcnt) ----
  for (int q = tid; q < HALO_QUADS; q += NTHREADS) {
    const int qc = q & (QPP - 1);
    const int p  = q / QPP;
    const int px = p % HALO_W;
    const int py = p / HALO_W;
    int gx = tx0 - 1 + px; gx = gx < 0 ? 0 : (gx > W_ - 1 ? W_ - 1 : gx);
    int gy = ty0 - 1 + py; gy = gy < 0 ? 0 : (gy > H_ - 1 ? H_ - 1 : gy);
    const float* gp = x + (((size_t)b * H_ + gy) * W_ + gx) * C_ + 4 * qc;
    async_copy_b128(gp, &tile[q]);
  }
  wait_async_all();     // drain this wave's ASYNCcnt
  __syncthreads();      // make LDS tile visible workgroup-wide

  const int c4 = tid & (QPP - 1);  // channel quad 0..7
  const int xo = tid / QPP;        // output x within tile 0..31

  // 16 weight quads for this thread's channels, held in VGPRs.
  float4 wq[KH_][KW_];
#pragma unroll
  for (int i = 0; i < KH_; ++i)
#pragma unroll
    for (int j = 0; j < KW_; ++j)
      wq[i][j] = *(const float4*)(w + ((i * KW_ + j) * C_) + 4 * c4);

  const bool interior = (tx0 >= 1) && (tx0 + TILE_W + 1 < W_) &&
                        (ty0 >= 1) && (ty0 + TILE_H + 1 < H_);
  if (interior)
    dilate_body<false>(tile, wq, out, b, ty0, tx0, xo, c4);
  else
    dilate_body<true>(tile, wq, out, b, ty0, tx0, xo, c4);
}

extern "C" void kernel_launch(void* const* d_in, const int* in_sizes, int n_in,
                              void* d_out, int out_size, void* d_ws, size_t ws_size,
                              hipStream_t stream) {
  const float* x = (const float*)d_in[0];
  const float* w = (const float*)d_in[1];
  float* out = (float*)d_out;
  dim3 grid(W_ / TILE_W, H_ / TILE_H, B_);   // 16 x 64 x 8 = 8192 blocks
  dilation2d_kernel<<<grid, dim3(NTHREADS), 0, stream>>>(x, w, out);
  (void)in_sizes; (void)n_in; (void)out_size; (void)d_ws; (void)ws_size;
}